// MultiSimilarityLoss_9225589752058
// MI455X (gfx1250) — compile-verified
//
#include <hip/hip_runtime.h>
#include <math.h>

typedef __attribute__((ext_vector_type(2))) float v2f;
typedef __attribute__((ext_vector_type(8))) float v8f;
typedef __attribute__((ext_vector_type(4))) int   v4i;

// Pointer types matching the async-load builtin's signature:
// param0 = v4i in address_space(1) (prints as "__device__" in HIP diags),
// param1 = v4i in address_space(3) ("__shared__").
typedef __attribute__((address_space(1))) v4i* ms_gptr_t;
typedef __attribute__((address_space(3))) v4i* ms_lptr_t;

#define MS_B    8192
#define MS_D    128
#define MS_NT   (MS_B / 16)      // 512 column tiles of width 16
#define MS_ROWB 128              // rows per block (8 waves x 16 rows)
#define MS_LDSC 132              // padded column stride in floats (bank-conflict-free)

__device__ const float MS_ALPHA  = 2.0f;
__device__ const float MS_BETA   = 50.0f;
__device__ const float MS_BASE   = 0.5f;
__device__ const float MS_MARGIN = 0.1f;

// ---- CDNA5 async global->LDS copy (ASYNCcnt path), builtin or asm fallback ----
__device__ __forceinline__ void ms_async_b128(const float* g, float* l) {
#if __has_builtin(__builtin_amdgcn_global_load_async_to_lds_b128)
    __builtin_amdgcn_global_load_async_to_lds_b128(
        (ms_gptr_t)(uintptr_t)g,
        (ms_lptr_t)(unsigned)(uintptr_t)l,   // low 32 bits = LDS offset
        /*offset=*/0, /*cpol=*/0);
#else
    // GV mode: vdst = LDS byte address, vaddr = 64-bit global address
    asm volatile("global_load_async_to_lds_b128 %0, %1, off"
                 :: "v"((unsigned)(uintptr_t)l), "v"(g)
                 : "memory");
#endif
}

__device__ __forceinline__ void ms_wait_async2() {
#if __has_builtin(__builtin_amdgcn_s_wait_asynccnt)
    __builtin_amdgcn_s_wait_asynccnt(2);
#else
    asm volatile("s_wait_asynccnt 0x2" ::: "memory");
#endif
}

__device__ __forceinline__ void ms_wait_async0() {
#if __has_builtin(__builtin_amdgcn_s_wait_asynccnt)
    __builtin_amdgcn_s_wait_asynccnt(0);
#else
    asm volatile("s_wait_asynccnt 0x0" ::: "memory");
#endif
}

// Stage column tile t (16 cols x 128 K of f32 = 8KB) into sB[buf].
// Each of the 256 threads copies two 16-byte chunks.
__device__ __forceinline__ void ms_stage_tile(const float* __restrict__ emb,
                                              float* __restrict__ sbuf,
                                              int t, int tid) {
#pragma unroll
    for (int h = 0; h < 2; ++h) {
        const int c   = tid + h * 256;     // chunk id 0..511
        const int col = c >> 5;            // 32 chunks (of 4 floats) per column
        const int k4  = (c & 31) * 4;
        ms_async_b128(emb + (size_t)(t * 16 + col) * MS_D + k4,
                      sbuf + col * MS_LDSC + k4);
    }
}

// One 16x16 f32 sim tile: 32 chained V_WMMA_F32_16X16X4_F32, B from LDS.
// Lane (nl, half): A frag a[j] = emb_row[4j+2*half .. +1]; B frag from the
// staged tile at column nl, same K slice (ds_load_b64, conflict-free).
__device__ __forceinline__ v8f ms_sim_tile_lds(const v2f* a, const float* bbase) {
    v8f c = {0.f, 0.f, 0.f, 0.f, 0.f, 0.f, 0.f, 0.f};
#pragma unroll
    for (int j = 0; j < 32; ++j) {
        v2f b = *(const v2f*)(bbase + 4 * j);
        c = __builtin_amdgcn_wmma_f32_16x16x4_f32(
            /*neg_a=*/false, a[j], /*neg_b=*/false, b,
            /*c_mod=*/(short)0, c, /*reuse_a=*/false, /*reuse_b=*/false);
    }
    return c;
}

// ---------------- Pass 1: per-row pos_min / neg_max ----------------
__global__ __launch_bounds__(256) void ms_pass1(const float* __restrict__ emb,
                                                const int* __restrict__ labels,
                                                float* __restrict__ pos_min,
                                                float* __restrict__ neg_max) {
    const int tid  = threadIdx.x;
    const int wave = tid >> 5;
    const int lane = tid & 31;
    const int nl   = lane & 15;
    const int half = lane >> 4;
    const int mw   = blockIdx.x * MS_ROWB + wave * 16;   // this wave's 16 rows
    const float FINF = __builtin_inff();

    __shared__ float sB[2][16 * MS_LDSC];

    // Preload this wave's A panel (row mw+nl, all 128 K values).
    v2f a[32];
    const float* arow = emb + (size_t)(mw + nl) * MS_D;
#pragma unroll
    for (int j = 0; j < 32; ++j)
        a[j] = *(const v2f*)(arow + 4 * j + 2 * half);

    int rlab[8], rowm[8];
#pragma unroll
    for (int v = 0; v < 8; ++v) {
        rowm[v] = mw + v + 8 * half;       // C VGPR v <-> row 8*half + v
        rlab[v] = labels[rowm[v]];
    }

    float pmin[8], nmax[8];
#pragma unroll
    for (int v = 0; v < 8; ++v) { pmin[v] = FINF; nmax[v] = -FINF; }

    ms_stage_tile(emb, sB[0], 0, tid);
    ms_stage_tile(emb, sB[1], 1, tid);

    for (int t = 0; t < MS_NT; ++t) {
        if (t + 1 < MS_NT) ms_wait_async2(); else ms_wait_async0();
        __syncthreads();                               // tile t fully in LDS

        const int col  = t * 16 + nl;
        const int clab = labels[col];
        const float* bbase = &sB[t & 1][nl * MS_LDSC + 2 * half];
        v8f c = ms_sim_tile_lds(a, bbase);
#pragma unroll
        for (int v = 0; v < 8; ++v) {
            const float s = c[v];
            if (rlab[v] == clab) { if (rowm[v] != col) pmin[v] = fminf(pmin[v], s); }
            else                 { nmax[v] = fmaxf(nmax[v], s); }
        }

        __syncthreads();                               // everyone done reading buf
        if (t + 2 < MS_NT) ms_stage_tile(emb, sB[t & 1], t + 2, tid);
    }

    // Reduce across the 16 column-lanes of each half; lanes 0/16 store.
#pragma unroll
    for (int v = 0; v < 8; ++v) {
#pragma unroll
        for (int mask = 8; mask >= 1; mask >>= 1) {
            pmin[v] = fminf(pmin[v], __shfl_xor(pmin[v], mask, 16));
            nmax[v] = fmaxf(nmax[v], __shfl_xor(nmax[v], mask, 16));
        }
    }
    if (nl == 0) {
#pragma unroll
        for (int v = 0; v < 8; ++v) {
            pos_min[rowm[v]] = pmin[v];
            neg_max[rowm[v]] = nmax[v];
        }
    }
}

// ---------------- Pass 2: mined exp-sums (recompute sim tiles) ----------------
__global__ __launch_bounds__(256) void ms_pass2(const float* __restrict__ emb,
                                                const int* __restrict__ labels,
                                                const float* __restrict__ pos_min,
                                                const float* __restrict__ neg_max,
                                                float* __restrict__ p_all,
                                                float* __restrict__ p_hard,
                                                float* __restrict__ n_all,
                                                float* __restrict__ n_hard,
                                                float* __restrict__ c_p,
                                                float* __restrict__ c_n) {
    const int tid  = threadIdx.x;
    const int wave = tid >> 5;
    const int lane = tid & 31;
    const int nl   = lane & 15;
    const int half = lane >> 4;
    const int mw   = blockIdx.x * MS_ROWB + wave * 16;

    __shared__ float sB[2][16 * MS_LDSC];

    v2f a[32];
    const float* arow = emb + (size_t)(mw + nl) * MS_D;
#pragma unroll
    for (int j = 0; j < 32; ++j)
        a[j] = *(const v2f*)(arow + 4 * j + 2 * half);

    int rlab[8], rowm[8];
    float thr_pmin[8], thr_nmax[8];
#pragma unroll
    for (int v = 0; v < 8; ++v) {
        rowm[v]     = mw + v + 8 * half;
        rlab[v]     = labels[rowm[v]];
        thr_pmin[v] = pos_min[rowm[v]];
        thr_nmax[v] = neg_max[rowm[v]];
    }

    // [v][0]=p_all [1]=p_hard [2]=n_all [3]=n_hard [4]=cnt_p_hard [5]=cnt_n_hard
    float acc[8][6];
#pragma unroll
    for (int v = 0; v < 8; ++v)
#pragma unroll
        for (int q = 0; q < 6; ++q) acc[v][q] = 0.f;

    ms_stage_tile(emb, sB[0], 0, tid);
    ms_stage_tile(emb, sB[1], 1, tid);

    for (int t = 0; t < MS_NT; ++t) {
        if (t + 1 < MS_NT) ms_wait_async2(); else ms_wait_async0();
        __syncthreads();

        const int col  = t * 16 + nl;
        const int clab = labels[col];
        const float* bbase = &sB[t & 1][nl * MS_LDSC + 2 * half];
        v8f c = ms_sim_tile_lds(a, bbase);
#pragma unroll
        for (int v = 0; v < 8; ++v) {
            const float s = c[v];
            if (rlab[v] == clab) {
                if (rowm[v] != col) {                 // positive pair
                    const float e = expf(-MS_ALPHA * (s - MS_BASE));
                    acc[v][0] += e;
                    if (s - MS_MARGIN < thr_nmax[v]) { acc[v][1] += e; acc[v][4] += 1.f; }
                }
            } else {                                  // negative pair
                const float e = expf(MS_BETA * (s - MS_BASE));
                acc[v][2] += e;
                if (s + MS_MARGIN > thr_pmin[v]) { acc[v][3] += e; acc[v][5] += 1.f; }
            }
        }

        __syncthreads();
        if (t + 2 < MS_NT) ms_stage_tile(emb, sB[t & 1], t + 2, tid);
    }

#pragma unroll
    for (int v = 0; v < 8; ++v)
#pragma unroll
        for (int q = 0; q < 6; ++q)
#pragma unroll
            for (int mask = 8; mask >= 1; mask >>= 1)
                acc[v][q] += __shfl_xor(acc[v][q], mask, 16);

    if (nl == 0) {
#pragma unroll
        for (int v = 0; v < 8; ++v) {
            const int row = rowm[v];
            p_all[row]  = acc[v][0];
            p_hard[row] = acc[v][1];
            n_all[row]  = acc[v][2];
            n_hard[row] = acc[v][3];
            c_p[row]    = acc[v][4];
            c_n[row]    = acc[v][5];
        }
    }
}

// ---------------- Pass 3: per-row loss + valid-row mean ----------------
__global__ __launch_bounds__(256) void ms_final(const float* __restrict__ pos_min,
                                                const float* __restrict__ neg_max,
                                                const float* __restrict__ p_all,
                                                const float* __restrict__ p_hard,
                                                const float* __restrict__ n_all,
                                                const float* __restrict__ n_hard,
                                                const float* __restrict__ c_p,
                                                const float* __restrict__ c_n,
                                                float* __restrict__ out) {
    const int tid = threadIdx.x;
    const float FINF = __builtin_inff();
    float sum = 0.f, cnt = 0.f;
    for (int r = tid; r < MS_B; r += 256) {
        const bool any_pos = pos_min[r] < FINF;    // finite <=> a positive existed
        const bool any_neg = neg_max[r] > -FINF;   // finite <=> a negative existed
        const float ps = (c_p[r] > 0.f) ? p_hard[r] : p_all[r];
        const float ns = (c_n[r] > 0.f) ? n_hard[r] : n_all[r];
        const float loss = log1pf(ps) / MS_ALPHA + log1pf(ns) / MS_BETA;
        if (any_pos && any_neg) { sum += loss; cnt += 1.f; }
    }
    __shared__ float ssum[256];
    __shared__ float scnt[256];
    ssum[tid] = sum;
    scnt[tid] = cnt;
    __syncthreads();
    for (int off = 128; off > 0; off >>= 1) {
        if (tid < off) {
            ssum[tid] += ssum[tid + off];
            scnt[tid] += scnt[tid + off];
        }
        __syncthreads();
    }
    if (tid == 0) out[0] = ssum[0] / fmaxf(scnt[0], 1.f);
}

extern "C" void kernel_launch(void* const* d_in, const int* in_sizes, int n_in,
                              void* d_out, int out_size, void* d_ws, size_t ws_size,
                              hipStream_t stream) {
    (void)in_sizes; (void)n_in; (void)out_size; (void)ws_size;
    const float* emb    = (const float*)d_in[0];
    const int*   labels = (const int*)d_in[1];
    float*       out    = (float*)d_out;

    float* w       = (float*)d_ws;     // 8 * 8192 floats = 256 KB scratch
    float* pos_min = w + 0 * MS_B;
    float* neg_max = w + 1 * MS_B;
    float* p_all   = w + 2 * MS_B;
    float* p_hard  = w + 3 * MS_B;
    float* n_all   = w + 4 * MS_B;
    float* n_hard  = w + 5 * MS_B;
    float* c_p     = w + 6 * MS_B;
    float* c_n     = w + 7 * MS_B;

    dim3 grid(MS_B / MS_ROWB), block(256);   // 64 blocks x 8 waves
    ms_pass1<<<grid, block, 0, stream>>>(emb, labels, pos_min, neg_max);
    ms_pass2<<<grid, block, 0, stream>>>(emb, labels, pos_min, neg_max,
                                         p_all, p_hard, n_all, n_hard, c_p, c_n);
    ms_final<<<1, block, 0, stream>>>(pos_min, neg_max, p_all, p_hard,
                                      n_all, n_hard, c_p, c_n, out);
}